// DARTS_82514911690825
// MI455X (gfx1250) — compile-verified
//
#include <hip/hip_runtime.h>
#include <cstdint>

// ---------------------------------------------------------------------------
// DARTS forward, MI455X (gfx1250, wave32).
// Streaming elementwise kernel: 12 B of HBM traffic per output row, ~70 VALU
// ops (8 of them TRANS). Bandwidth/TRANS bound -> no WMMA (2x2 matmul, K=2).
// Uses CDNA5 async global->LDS pipeline (ASYNCcnt) to hide HBM latency.
// ---------------------------------------------------------------------------

#define BLOCK 256
#define DEPTH 4     // async pipeline stages (per-lane float4 slots)
#define GRID  1024

#define EPSF 1e-10f
#define YTHF 22026.465794806718f   // exp(10) == clip threshold y_th

typedef float v2f __attribute__((ext_vector_type(2)));   // native vec for NT store

// ---------------- setup: fold all parameters into 56 floats ----------------
__global__ void darts_setup(const float* __restrict__ alphas,
                            const float* __restrict__ op_w,
                            const float* __restrict__ op_b,
                            const float* __restrict__ W1,
                            const float* __restrict__ b1,
                            const float* __restrict__ wo,
                            const float* __restrict__ bo,
                            float* __restrict__ C) {
  if (blockIdx.x == 0 && threadIdx.x == 0) {
    // softmax(alphas / 1.0) per input row (max-subtracted, like jax.nn.softmax)
    for (int i = 0; i < 2; ++i) {
      float m = alphas[i * 8];
      for (int k = 1; k < 8; ++k) m = fmaxf(m, alphas[i * 8 + k]);
      float e[8], s = 0.0f;
      for (int k = 0; k < 8; ++k) { e[k] = __expf(alphas[i * 8 + k] - m); s += e[k]; }
      float inv = 1.0f / s;
      for (int k = 0; k < 8; ++k) C[i * 8 + k] = e[k] * inv;     // C[0..15]  = w
    }
    for (int t = 0; t < 16; ++t) { C[16 + t] = op_w[t]; C[32 + t] = op_b[t]; }
    for (int t = 0; t < 4; ++t) C[48 + t] = W1[t];               // row-major 2x2
    C[52] = b1[0]; C[53] = b1[1];
    C[54] = wo[0]; C[55] = bo[0];
  }
}

// ------------- per-hidden-unit mixture: sum_k w_k*clip(ow_k*f_k+ob_k) -------
__device__ __forceinline__ float eval7(float h, const float* w,
                                       const float* ow, const float* ob) {
  float f1 = h;                                  // linear
  float f2 = h * h;                              // power_two
  float f3 = f2 * h;                             // power_three
  float f4 = __expf(fminf(h, 10.0f));            // safe exp  -> v_exp_f32
  float f5 = __logf(fabsf(h) + EPSF);            // safe ln   -> v_log_f32
  float d  = h + (h >= 0.0f ? EPSF : -EPSF);     // safe reciprocal denom
  float r  = __builtin_amdgcn_rcpf(d);           // v_rcp_f32 + 1 Newton step
  r = __builtin_fmaf(__builtin_fmaf(-d, r, 1.0f), r, r);
  float f7 = __sinf(h);                          // v_sin_f32 (scaled by header)
  float f[7] = { f1, f2, f3, f4, f5, r, f7 };
  float acc = 0.0f;
#pragma unroll
  for (int k = 0; k < 7; ++k) {
    float v = __builtin_fmaf(ow[k], f[k], ob[k]);
    v = fminf(fmaxf(v, -YTHF), YTHF);            // clip to +/- exp(10)
    acc = __builtin_fmaf(w[k], v, acc);
  }
  return acc;
}

// ----------------------------- main streaming kernel ------------------------
__global__ __launch_bounds__(BLOCK) void darts_main(const float* __restrict__ x,
                                                    const float* __restrict__ C,
                                                    float* __restrict__ out,
                                                    unsigned total4) {
  __shared__ float4 tile[DEPTH][BLOCK];   // 16 KB per block
  __shared__ float  sC[56];

  const unsigned tid  = threadIdx.x;
  const unsigned T    = GRID * BLOCK;                 // total threads
  const unsigned gtid = blockIdx.x * BLOCK + tid;
  const unsigned niter = (total4 + T - 1) / T;

  if (tid < 56) sC[tid] = C[tid];                     // one L2 read per block
  __syncthreads();

  // register-resident constants (k = 1..7; 'none' op is identically zero)
  float w[2][7], ow[2][7], ob[2][7];
#pragma unroll
  for (int i = 0; i < 2; ++i)
#pragma unroll
    for (int k = 0; k < 7; ++k) {
      w[i][k]  = sC[i * 8 + 1 + k];
      ow[i][k] = sC[16 + i * 8 + 1 + k];
      ob[i][k] = sC[32 + i * 8 + 1 + k];
    }
  const float w00 = sC[48], w01 = sC[49], w10 = sC[50], w11 = sC[51];
  const float bb0 = sC[52], bb1 = sC[53], woc = sC[54], boc = sC[55];

  // ---- depth-4 async global->LDS pipeline (per-lane, no cross-lane deps) ---
  const uint64_t xbase = (uint64_t)(uintptr_t)x;
  auto issue = [&](unsigned it) {
    unsigned idx = gtid + it * T;
    if (idx > total4 - 1u) idx = total4 - 1u;         // clamped dummy prefetch
    uint64_t ga = xbase + (uint64_t)idx * 16u;
    uint32_t la = (uint32_t)(uintptr_t)&tile[it & (DEPTH - 1)][tid];
    asm volatile("global_load_async_to_lds_b128 %0, %1, off"
                 :: "v"(la), "v"(ga) : "memory");     // tracked by ASYNCcnt
  };
#pragma unroll
  for (unsigned s = 0; s < DEPTH; ++s) issue(s);      // prologue: 4 in flight

  v2f* __restrict__ out2 = (v2f*)out;
  for (unsigned it = 0; it < niter; ++it) {
    // 4 outstanding, in-order completion: <=3 left  =>  oldest stage landed
    asm volatile("s_wait_asynccnt 0x3" ::: "memory");
    float4 v = tile[it & (DEPTH - 1)][tid];           // ds_load_b128, no conflicts
    unsigned idx = gtid + it * T;                     // float4 idx = 2 rows

    // h = x @ W1^T + b1 for both rows in this float4
    float h00 = __builtin_fmaf(v.x, w00, __builtin_fmaf(v.y, w01, bb0));
    float h01 = __builtin_fmaf(v.x, w10, __builtin_fmaf(v.y, w11, bb1));
    float h10 = __builtin_fmaf(v.z, w00, __builtin_fmaf(v.w, w01, bb0));
    float h11 = __builtin_fmaf(v.z, w10, __builtin_fmaf(v.w, w11, bb1));

    float a0 = eval7(h00, w[0], ow[0], ob[0]) + eval7(h01, w[1], ow[1], ob[1]);
    float a1 = eval7(h10, w[0], ow[0], ob[0]) + eval7(h11, w[1], ow[1], ob[1]);

    v2f o;
    o.x = __builtin_fmaf(a0, woc, boc);
    o.y = __builtin_fmaf(a1, woc, boc);
    if (idx < total4) __builtin_nontemporal_store(o, &out2[idx]);  // read-never data

    // ensure the ds_load of this stage fully landed in VGPRs before the async
    // write below overwrites the same LDS slot (WAR at the LDS level)
    asm volatile("s_wait_dscnt 0x0" ::: "memory");
    issue(it + DEPTH);                                // refill pipeline
  }
  // leftover dummy async loads drained by implicit wait-idle at s_endpgm
}

// --------------------------------- launcher ---------------------------------
extern "C" void kernel_launch(void* const* d_in, const int* in_sizes, int n_in,
                              void* d_out, int out_size, void* d_ws, size_t ws_size,
                              hipStream_t stream) {
  const float* x      = (const float*)d_in[0];
  const float* W1     = (const float*)d_in[1];
  const float* b1     = (const float*)d_in[2];
  const float* alphas = (const float*)d_in[3];
  const float* op_w   = (const float*)d_in[4];
  const float* op_b   = (const float*)d_in[5];
  const float* wo     = (const float*)d_in[6];
  const float* bo     = (const float*)d_in[7];
  float* out = (float*)d_out;
  float* C   = (float*)d_ws;                  // 56 floats of folded constants

  unsigned total_x = (unsigned)in_sizes[0];   // B * 2 floats
  unsigned total4  = total_x / 4u;            // float4 units (2 rows each)

  darts_setup<<<1, 64, 0, stream>>>(alphas, op_w, op_b, W1, b1, wo, bo, C);
  darts_main<<<GRID, BLOCK, 0, stream>>>(x, C, out, total4);
}